// CircularCorrelation_10075993276568
// MI455X (gfx1250) — compile-verified
//
#include <hip/hip_runtime.h>

#define N 4096
#define K1_THREADS 1024
#define MAX_ROWS 64   // rows-per-block cap for pass1 (NB >= 64)

typedef float v2f __attribute__((ext_vector_type(2)));
typedef float v8f __attribute__((ext_vector_type(8)));

// ---------------------------------------------------------------------------
// Pass 1: stream km1/km2 exactly once (128 MiB total -> HBM-bound).
// Per element: 1 FMA into per-thread column partial (t = km^T o) and
// 1 FMA into per-row circulant dot (d[j] = sum_k key[(k-j)%N]*km[j,k]).
// Deterministic reductions only (no fp atomics).
// ---------------------------------------------------------------------------
__global__ __launch_bounds__(K1_THREADS) void pass1_stream(
    const float* __restrict__ key,
    const float* __restrict__ km1, const float* __restrict__ o1,
    const float* __restrict__ km2, const float* __restrict__ o2,
    float* __restrict__ p1, float* __restrict__ p2,
    float* __restrict__ d1, float* __restrict__ d2,
    int rowsPerBlock)
{
    __shared__ float keyL[N];
    __shared__ float d1w[MAX_ROWS][33];   // [row][wave], padded: conflict-free
    __shared__ float d2w[MAX_ROWS][33];

    const int tid  = threadIdx.x;
    const int lane = tid & 31;
    const int wave = tid >> 5;

    for (int i = tid; i < N; i += K1_THREADS) keyL[i] = key[i];
    __syncthreads();

    const int rowBase = blockIdx.x * rowsPerBlock;

    float4 t1a = make_float4(0.f, 0.f, 0.f, 0.f);
    float4 t2a = make_float4(0.f, 0.f, 0.f, 0.f);

    for (int r = 0; r < rowsPerBlock; ++r) {
        const int j = rowBase + r;
        const float o1j = o1[j];              // uniform -> scalar load
        const float o2j = o2[j];
        const float4 a = ((const float4*)(km1 + (size_t)j * N))[tid]; // b128 coalesced
        const float4 b = ((const float4*)(km2 + (size_t)j * N))[tid];

        const int kb = 4 * tid - j + N;       // circulant index base, non-negative
        const float w0 = keyL[(kb    ) & (N - 1)];
        const float w1 = keyL[(kb + 1) & (N - 1)];
        const float w2 = keyL[(kb + 2) & (N - 1)];
        const float w3 = keyL[(kb + 3) & (N - 1)];

        t1a.x += o1j * a.x; t1a.y += o1j * a.y; t1a.z += o1j * a.z; t1a.w += o1j * a.w;
        t2a.x += o2j * b.x; t2a.y += o2j * b.y; t2a.z += o2j * b.z; t2a.w += o2j * b.w;

        float d1p = w0 * a.x + w1 * a.y + w2 * a.z + w3 * a.w;
        float d2p = w0 * b.x + w1 * b.y + w2 * b.z + w3 * b.w;

        // wave32 butterfly reduction (deterministic)
        for (int off = 16; off > 0; off >>= 1) {
            d1p += __shfl_xor(d1p, off, 32);
            d2p += __shfl_xor(d2p, off, 32);
        }
        if (lane == 0) { d1w[r][wave] = d1p; d2w[r][wave] = d2p; }
    }
    __syncthreads();

    // finish per-row dots: fixed-order sum over the 32 waves
    for (int t = tid; t < rowsPerBlock; t += K1_THREADS) {
        float s1 = 0.f, s2 = 0.f;
        for (int w = 0; w < 32; ++w) { s1 += d1w[t][w]; s2 += d2w[t][w]; }
        d1[rowBase + t] = s1;
        d2[rowBase + t] = s2;
    }

    // per-block column partials of km^T o (thread owns cols 4*tid..4*tid+3)
    ((float4*)(p1 + (size_t)blockIdx.x * N))[tid] = t1a;
    ((float4*)(p2 + (size_t)blockIdx.x * N))[tid] = t2a;
}

// ---------------------------------------------------------------------------
// Pass 2: v[k] = x[k] - sum_b p1[b][k] - sum_b p2[b][k]
// ---------------------------------------------------------------------------
__global__ void pass2_reduce(const float* __restrict__ x,
                             const float* __restrict__ p1,
                             const float* __restrict__ p2,
                             float* __restrict__ v, int nb)
{
    const int k = blockIdx.x * blockDim.x + threadIdx.x;
    float s = 0.f;
    for (int b = 0; b < nb; ++b)
        s += p1[(size_t)b * N + k] + p2[(size_t)b * N + k];
    v[k] = x[k] - s;
}

// ---------------------------------------------------------------------------
// Pass 3: c = K @ v (circular correlation) via V_WMMA_F32_16X16X4_F32,
// then result[j] = (c[j] + d1[j]*o1[j] + d2[j]*o2[j]) / d1[j]; out[N] = d2[N-1].
//
// Per wave: one 16-row output tile (j0..j0+15). f32 A layout (16x4, M x K):
//   lane L (m = L&15, g = L>>4), VGPR p  ->  A[m, K = p + 2g]
// B (4x16, K x N) broadcasts v across all 16 columns, so every column of the
// 16x16 accumulator equals the wanted c-tile. 4 independent acc chains.
// ---------------------------------------------------------------------------
__global__ __launch_bounds__(128) void pass3_wmma(
    const float* __restrict__ key, const float* __restrict__ v,
    const float* __restrict__ d1, const float* __restrict__ d2,
    const float* __restrict__ o1, const float* __restrict__ o2,
    float* __restrict__ out)
{
    __shared__ float keyL[N];
    __shared__ float vL[N];
    for (int i = threadIdx.x; i < N; i += 128) { keyL[i] = key[i]; vL[i] = v[i]; }
    __syncthreads();

    const int wave = threadIdx.x >> 5;
    const int lane = threadIdx.x & 31;
    const int j0 = (blockIdx.x * 4 + wave) * 16;
    const int m = lane & 15;
    const int g = lane >> 4;
    const int abase = 2 * g - j0 - m + 2 * N;   // keep index positive before masking

    v8f acc0 = {}, acc1 = {}, acc2 = {}, acc3 = {};

    for (int k0 = 0; k0 < N; k0 += 16) {
        v2f a, b;
#define CC_STEP(ACC, OFF)                                                     \
        a.x = keyL[(k0 + (OFF) + abase    ) & (N - 1)];                       \
        a.y = keyL[(k0 + (OFF) + abase + 1) & (N - 1)];                       \
        b.x = vL[k0 + (OFF) + 2 * g];                                         \
        b.y = vL[k0 + (OFF) + 2 * g + 1];                                     \
        ACC = __builtin_amdgcn_wmma_f32_16x16x4_f32(false, a, false, b,       \
                                                    (short)0, ACC, false, false);
        CC_STEP(acc0, 0)
        CC_STEP(acc1, 4)
        CC_STEP(acc2, 8)
        CC_STEP(acc3, 12)
#undef CC_STEP
    }

    v8f c = (acc0 + acc1) + (acc2 + acc3);

    // C layout: VGPR p, lanes 0-15 -> M=p ; lanes 16-31 -> M=p+8 (cols identical)
    if (m == 0) {
        for (int p = 0; p < 8; ++p) {
            const int j = j0 + 8 * g + p;
            const float den = d1[j];
            out[j] = (c[p] + den * o1[j] + d2[j] * o2[j]) / den;
        }
    }
    if (blockIdx.x == 0 && threadIdx.x == 0) out[N] = d2[N - 1];  // redun3
}

// ---------------------------------------------------------------------------
extern "C" void kernel_launch(void* const* d_in, const int* in_sizes, int n_in,
                              void* d_out, int out_size, void* d_ws, size_t ws_size,
                              hipStream_t stream)
{
    (void)in_sizes; (void)n_in; (void)out_size;

    const float* key = (const float*)d_in[0];
    const float* x   = (const float*)d_in[1];
    const float* km1 = (const float*)d_in[2];
    const float* o1  = (const float*)d_in[3];
    const float* km2 = (const float*)d_in[4];
    const float* o2  = (const float*)d_in[5];
    float* out = (float*)d_out;

    int NB = 256;  // row-tiles (blocks) for pass1 -> 16 rows/block
    if (((size_t)2 * NB * N + 3 * N) * sizeof(float) > ws_size) NB = 64;

    float* p1 = (float*)d_ws;
    float* p2 = p1 + (size_t)NB * N;
    float* d1 = p2 + (size_t)NB * N;
    float* d2 = d1 + N;
    float* v  = d2 + N;

    const int rowsPerBlock = N / NB;

    hipLaunchKernelGGL(pass1_stream, dim3(NB), dim3(K1_THREADS), 0, stream,
                       key, km1, o1, km2, o2, p1, p2, d1, d2, rowsPerBlock);
    hipLaunchKernelGGL(pass2_reduce, dim3(N / 256), dim3(256), 0, stream,
                       x, p1, p2, v, NB);
    hipLaunchKernelGGL(pass3_wmma, dim3(N / 64), dim3(128), 0, stream,
                       key, v, d1, d2, o1, o2, out);
}